// UBG_GCN_50697793962365
// MI455X (gfx1250) — compile-verified
//
#include <hip/hip_runtime.h>

typedef __attribute__((ext_vector_type(2))) float v2f;
typedef __attribute__((ext_vector_type(8))) float v8f;

#if defined(__has_builtin) && __has_builtin(__builtin_amdgcn_sched_barrier)
#define SCHED_PIN() __builtin_amdgcn_sched_barrier(0)
#else
#define SCHED_PIN()
#endif

// -------------------------------------------------------------------------
// Zero-init (graph-capture safe, no hipMemset)
// -------------------------------------------------------------------------
__global__ void UBG_zero_f32(float* __restrict__ p, size_t n) {
    size_t i = (size_t)blockIdx.x * blockDim.x + threadIdx.x;
    size_t stride = (size_t)gridDim.x * blockDim.x;
    for (; i < n; i += stride) p[i] = 0.0f;
}

// -------------------------------------------------------------------------
// SpMM over row-sorted COO: out[r,:] += vals[e] * x[cols[e],:]
// One block of D threads (thread = feature) walks a chunk of edges serially,
// accumulating in a register while the row is unchanged; flushes with a
// global f32 atomic on row change and at chunk end. Gathered x rows are
// coalesced 512B transactions and (51MB) L2-resident on MI455X (192MB L2).
// -------------------------------------------------------------------------
template<int D>
__global__ __launch_bounds__(D) void UBG_spmm_sorted(
    const float* __restrict__ vals, const int* __restrict__ rows,
    const int* __restrict__ cols, const float* __restrict__ x,
    float* __restrict__ out, int E, int edges_per_block) {
    const int f = threadIdx.x;
    long e0 = (long)blockIdx.x * edges_per_block;
    long e1 = e0 + edges_per_block;
    if (e1 > (long)E) e1 = (long)E;
    if (e0 >= e1) return;

    int cur = rows[e0];
    float acc = 0.0f;
    for (long e = e0; e < e1; ++e) {
        const int r = rows[e];
        if (r != cur) {
            atomicAdd(&out[(size_t)cur * D + f], acc);
            acc = 0.0f;
            cur = r;
        }
        acc = fmaf(vals[e], x[(size_t)cols[e] * D + f], acc);
    }
    atomicAdd(&out[(size_t)cur * D + f], acc);
}

// -------------------------------------------------------------------------
// Dense GEMM via V_WMMA_F32_16X16X4_F32 (bit-exact fp32 matrix pipe).
//   out[M,N] = [relu]( H[M,K] @ W[K,N] + bias[N] )
// 256 threads = 8 waves; each wave owns one 16-row M-tile and all N/16
// column tiles.
//
// W is staged in LDS *K-pair interleaved*: LDS entry p = (k/2)*N + n is the
// v2f pair (W[k][n], W[k+1][n]) for even k. Each WMMA B fragment is then a
// single 8B-aligned ds_load_b64 directly into the even-aligned VGPR pair the
// matrix pipe requires -- this removes the v_dual_mov re-pairing storm that
// ds_load_2addr produced in earlier rounds. Lane's pair row = kk/2 + half.
//
// A fragment is software-pipelined one k-step ahead; sched_barrier(0) pins
// the prefetch issue point so the scheduler cannot sink the global_load to
// its use (round-2 asm showed global_load_b64 ; s_wait_loadcnt 0 ; v_wmma).
//
// ISA fragment layouts (CDNA5 ISA):
//   A (16x4, 2 VGPRs): lanes 0-15 -> M=lane, {K=kk, kk+1};
//                      lanes 16-31 -> M=lane-16, {K=kk+2, kk+3}
//   B (4x16, 2 VGPRs): mirrored K striping, N = lane&15 per VGPR
//   C/D (16x16, 8 VGPRs): VGPR i -> row i (lanes 0-15) / row i+8 (16-31),
//                          col = lane&15
// -------------------------------------------------------------------------
template<int K, int N, bool RELU>
__global__ __launch_bounds__(256) void UBG_gemm_wmma_f32(
    const float* __restrict__ H, const float* __restrict__ W,
    const float* __restrict__ bias, float* __restrict__ out, int M) {
    __shared__ v2f Wl[(K / 2) * N];     // K-pair interleaved copy of W
    {
        float* Wf = (float*)Wl;
        for (int i = threadIdx.x; i < K * N; i += 256) {
            const int k = i / N;        // N is a power of two -> shifts
            const int n = i % N;
            Wf[((k >> 1) * N + n) * 2 + (k & 1)] = W[i];
        }
    }
    __syncthreads();

    const int wave = threadIdx.x >> 5;
    const int lane = threadIdx.x & 31;
    const int half = lane >> 4;   // which K pair this lane holds
    const int l15  = lane & 15;
    const int m0 = (blockIdx.x * 8 + wave) * 16;
    if (m0 >= M) return;          // wave-uniform exit: EXEC all-ones for WMMA

    constexpr int NT = N / 16;
    v8f c[NT] = {};

    // Row pointer pre-offset by this lane's K-pair (2*half floats -> 8B align).
    const float* hrow = H + (size_t)(m0 + l15) * K + 2 * half;
    // This lane's B pair rows start at kk/2 + half.
    const v2f* Wp = Wl + (size_t)half * N + l15;

    v2f a_cur = *(const v2f*)(hrow);            // fragment for kk = 0
    for (int kk = 0; kk < K; kk += 4) {
        // Prefetch next A fragment (clamped: last iter redundantly reloads),
        // then pin the issue point so it stays a full chain ahead of its use.
        const int knext = (kk + 4 < K) ? (kk + 4) : kk;
        v2f a_nxt = *(const v2f*)(hrow + knext);
        SCHED_PIN();

        const v2f* Wrow = Wp + (kk >> 1) * N;   // pair (W[ka][n], W[ka+1][n])
#pragma unroll
        for (int t = 0; t < NT; ++t) {
            v2f b = Wrow[t * 16];
            // (neg_a, A, neg_b, B, c_mod, C, reuse_a, reuse_b)
            c[t] = __builtin_amdgcn_wmma_f32_16x16x4_f32(
                false, a_cur, false, b, (short)0, c[t], false, false);
        }
        a_cur = a_nxt;
    }

#pragma unroll
    for (int t = 0; t < NT; ++t) {
        const int n = t * 16 + l15;
        const float bv = bias[n];
#pragma unroll
        for (int i = 0; i < 8; ++i) {
            const int m = m0 + i + 8 * half;
            float v = c[t][i] + bv;
            if (RELU) v = fmaxf(v, 0.0f);
            out[(size_t)m * N + n] = v;
        }
    }
}

// -------------------------------------------------------------------------
// Launch: zero -> spmm1 -> gemm1(relu) -> zero -> spmm2 -> gemm2
// -------------------------------------------------------------------------
extern "C" void kernel_launch(void* const* d_in, const int* in_sizes, int n_in,
                              void* d_out, int out_size, void* d_ws, size_t ws_size,
                              hipStream_t stream) {
    const float* x    = (const float*)d_in[0];
    const float* vals = (const float*)d_in[1];
    const int*   rows = (const int*)d_in[2];
    const int*   cols = (const int*)d_in[3];
    const float* W1   = (const float*)d_in[4];
    const float* b1   = (const float*)d_in[5];
    const float* W2   = (const float*)d_in[6];
    const float* b2   = (const float*)d_in[7];
    float* out = (float*)d_out;

    const int Nn = in_sizes[0] / 128;   // 100000 nodes
    const int E  = in_sizes[1];         // 1.6M edges
    const size_t nh = (size_t)Nn * 128;

    float* bufA = (float*)d_ws;         // h0 / h2 (51.2 MB)
    float* bufB = bufA + nh;            // h1      (51.2 MB)

    const int EPB = 128;                               // edges per block
    const int spmm_blocks = (E + EPB - 1) / EPB;       // 12500
    const int mtiles = (Nn + 15) / 16;                 // 6250 (exact)
    const int gemm_blocks = (mtiles + 7) / 8;          // 782

    UBG_zero_f32<<<2048, 256, 0, stream>>>(bufA, nh);
    UBG_spmm_sorted<128><<<spmm_blocks, 128, 0, stream>>>(
        vals, rows, cols, x, bufA, E, EPB);
    UBG_gemm_wmma_f32<128, 128, true><<<gemm_blocks, 256, 0, stream>>>(
        bufA, W1, b1, bufB, Nn);

    UBG_zero_f32<<<2048, 256, 0, stream>>>(bufA, nh);
    UBG_spmm_sorted<128><<<spmm_blocks, 128, 0, stream>>>(
        vals, rows, cols, bufB, bufA, E, EPB);
    UBG_gemm_wmma_f32<128, 64, false><<<gemm_blocks, 256, 0, stream>>>(
        bufA, W2, b2, out, Nn);

    (void)n_in; (void)out_size; (void)ws_size;
}